// GATLayer_75788992905466
// MI455X (gfx1250) — compile-verified
//
#include <hip/hip_runtime.h>

// ---------------------------------------------------------------------------
// GAT layer forward for gfx1250 (MI455X).
//   x[N,128] @ W[128,128] -> h   (fp32 WMMA 16x16x4; GEMM is bandwidth-bound:
//   ~51 MB @ 23.3 TB/s ~ 2us, so full-precision WMMA is the right choice)
//   A tile staged into LDS via the CDNA5 Tensor Data Mover (TENSORcnt path),
//   then per-node attention scalars and a 3-pass segment softmax + scatter.
// ---------------------------------------------------------------------------

typedef __attribute__((ext_vector_type(2))) float v2f;
typedef __attribute__((ext_vector_type(8))) float v8f;
typedef __attribute__((ext_vector_type(4))) unsigned int v4u;
typedef __attribute__((ext_vector_type(8))) int v8i;
typedef __attribute__((ext_vector_type(4))) int v4i;

__device__ __forceinline__ float lrelu(float x) { return x > 0.f ? x : 0.2f * x; }

// float atomic max via signed-int max / unsigned-int min trick
__device__ __forceinline__ void atomicMaxF(float* addr, float v) {
  if (v >= 0.f) atomicMax((int*)addr, __float_as_int(v));
  else          atomicMin((unsigned int*)addr, __float_as_uint(v));
}

#if __has_builtin(__builtin_amdgcn_tensor_load_to_lds) && \
    __has_builtin(__builtin_amdgcn_s_wait_tensorcnt)
#define GAT_USE_TDM 1
#endif

// ---------------------------------------------------------------------------
// K1: h = x @ W.  One block = 256 threads = 8 waves, covers 16 rows x 128 cols.
// Wave w computes the 16x16 tile at columns [16w, 16w+16).
// A tile (16x128 of x, 8 KB) staged in LDS by the TDM; 32 unrolled
// v_wmma_f32_16x16x4_f32 per wave.  N must be a multiple of 16 (50000=3125*16).
// ---------------------------------------------------------------------------
__global__ __launch_bounds__(256) void gat_gemm(const float* __restrict__ x,
                                                const float* __restrict__ W,
                                                float* __restrict__ h) {
  __shared__ float As[16 * 128];  // 8 KB

  const int tid  = threadIdx.x;
  const int lane = tid & 31;
  const int wave = tid >> 5;
  const int row0 = blockIdx.x * 16;

#ifdef GAT_USE_TDM
  // Tensor Data Mover: DMA the 16x128 fp32 tile global->LDS, one D# per block.
  if (wave == 0) {
    const unsigned long long ga =
        (unsigned long long)(const void*)(x + (size_t)row0 * 128);
    // Flat LDS address: low 32 bits are the LDS byte offset (ISA 10.2).
    const unsigned int lds = (unsigned int)(unsigned long long)(uintptr_t)As;
    v4u g0;
    g0.x = 1u;                                   // count=1 (valid), load, no gather
    g0.y = lds;                                  // D#.lds_addr
    g0.z = (unsigned int)ga;                     // D#.global_addr[31:0]
    g0.w = (unsigned int)(ga >> 32) | (2u << 30);// global_addr[56:32] | type=2
    v8i g1;
    g1[0] = (2 << 16);    // workgroup_mask=0 | data_size=2 (4 bytes) | flags=0
    g1[1] = (128 << 16);  // atomic_barrier_addr=0 | tensor_dim0 = 128 (lo16)
    g1[2] = (16 << 16);   // tensor_dim0 hi=0      | tensor_dim1 = 16  (lo16)
    g1[3] = (128 << 16);  // tensor_dim1 hi=0      | tile_dim0  = 128
    g1[4] = 16;           // tile_dim1 = 16        | tile_dim2  = 0
    g1[5] = 128;          // tensor_dim0_stride = 128 elements (lo32)
    g1[6] = 0;            // stride hi | tensor_dim1_stride lo (unused, 2-D)
    g1[7] = 0;
    v4i g2 = {0, 0, 0, 0};            // 2-D tile: groups 2/3 unused
    v4i g3 = {0, 0, 0, 0};
    v8i g4 = {0, 0, 0, 0, 0, 0, 0, 0};// extra group (6-arg toolchain form)
    __builtin_amdgcn_tensor_load_to_lds(g0, g1, g2, g3, g4, 0);
    __builtin_amdgcn_s_wait_tensorcnt(0);
  }
  __syncthreads();
#else
  // Fallback: cooperative 16x128 tile load, 512 float4, 2 per thread.
  const float4* xg = (const float4*)(x + (size_t)row0 * 128);
  float4* As4 = (float4*)As;
  As4[tid]       = xg[tid];
  As4[tid + 256] = xg[tid + 256];
  __syncthreads();
#endif

  const int n0  = wave * 16;        // column base of this wave's tile
  const int mlo = lane & 15;
  const int khi = lane >> 4;        // 0: K pair {0,1}; 1: K pair {2,3}

  v8f acc = {};
#pragma unroll
  for (int k = 0; k < 32; ++k) {
    const int kk = 4 * k + 2 * khi;
    // A frag: lane<16 holds A[M=lane][kk..kk+1]; lane>=16 A[M=lane-16][kk..kk+1]
    v2f a = *(const v2f*)&As[mlo * 128 + kk];
    // B frag: lane holds B[K=kk][n0+mlo] and B[K=kk+1][n0+mlo]
    v2f b;
    b.x = W[(size_t)kk * 128 + n0 + mlo];
    b.y = W[(size_t)(kk + 1) * 128 + n0 + mlo];
    acc = __builtin_amdgcn_wmma_f32_16x16x4_f32(false, a, false, b,
                                                (short)0, acc, false, false);
  }

  // D layout: VGPR r -> row r (lanes 0-15) / row 8+r (lanes 16-31), col = n0+mlo
  float* hp = h + (size_t)(row0 + 8 * khi) * 128 + n0 + mlo;
#pragma unroll
  for (int r = 0; r < 8; ++r) hp[(size_t)r * 128] = acc[r];
}

// ---------------------------------------------------------------------------
// K2: per-node attention scalars + state init. One wave (32 lanes) per node.
//   a_src[n,h] = sum_c h[n,h,c]*att_src[h,c]   (att flat[128] matches h cols)
//   e_max init = self-loop logit;  denom init = 0;  out init = bias.
// ---------------------------------------------------------------------------
__global__ __launch_bounds__(256) void gat_node(const float* __restrict__ h,
                                                const float* __restrict__ att_src,
                                                const float* __restrict__ att_dst,
                                                const float* __restrict__ bias,
                                                float* __restrict__ a_src,
                                                float* __restrict__ a_dst,
                                                float* __restrict__ e_max,
                                                float* __restrict__ denom,
                                                float* __restrict__ out, int N) {
  const int lane = threadIdx.x & 31;
  const int n = blockIdx.x * 8 + (threadIdx.x >> 5);
  if (n >= N) return;

  const float* hn = h + (size_t)n * 128;
  float h0 = hn[lane], h1 = hn[lane + 32], h2 = hn[lane + 64], h3 = hn[lane + 96];
  float s0 = h0 * att_src[lane]      + h1 * att_src[lane + 32];
  float s1 = h2 * att_src[lane + 64] + h3 * att_src[lane + 96];
  float d0 = h0 * att_dst[lane]      + h1 * att_dst[lane + 32];
  float d1 = h2 * att_dst[lane + 64] + h3 * att_dst[lane + 96];
#pragma unroll
  for (int off = 16; off; off >>= 1) {
    s0 += __shfl_xor(s0, off, 32);
    s1 += __shfl_xor(s1, off, 32);
    d0 += __shfl_xor(d0, off, 32);
    d1 += __shfl_xor(d1, off, 32);
  }

  out[(size_t)n * 64 + lane]      = bias[lane];
  out[(size_t)n * 64 + lane + 32] = bias[lane + 32];

  if (lane == 0) {
    a_src[n * 2] = s0;  a_src[n * 2 + 1] = s1;
    a_dst[n * 2] = d0;  a_dst[n * 2 + 1] = d1;
    e_max[n * 2]     = lrelu(s0 + d0);   // self-loop seeds the segment max
    e_max[n * 2 + 1] = lrelu(s1 + d1);
    denom[n * 2] = 0.f;  denom[n * 2 + 1] = 0.f;
  }
}

// ---------------------------------------------------------------------------
// K3: segment max over incoming edges (self-loop already seeded).
// ---------------------------------------------------------------------------
__global__ __launch_bounds__(256) void gat_edge_max(const int* __restrict__ ei, int E,
                                                    const float* __restrict__ a_src,
                                                    const float* __restrict__ a_dst,
                                                    float* __restrict__ e_max) {
  const int e = blockIdx.x * 256 + threadIdx.x;
  if (e >= E) return;
  const int s = ei[e], d = ei[E + e];
  atomicMaxF(&e_max[d * 2],     lrelu(a_src[s * 2]     + a_dst[d * 2]));
  atomicMaxF(&e_max[d * 2 + 1], lrelu(a_src[s * 2 + 1] + a_dst[d * 2 + 1]));
}

// ---------------------------------------------------------------------------
// K4: segment sum of exp(e - max).  i in [0,E) = real edges, [E,E+N) = self loops.
// ---------------------------------------------------------------------------
__global__ __launch_bounds__(256) void gat_edge_sum(const int* __restrict__ ei, int E, int N,
                                                    const float* __restrict__ a_src,
                                                    const float* __restrict__ a_dst,
                                                    const float* __restrict__ e_max,
                                                    float* __restrict__ denom) {
  const int i = blockIdx.x * 256 + threadIdx.x;
  if (i >= E + N) return;
  int s, d;
  if (i < E) { s = ei[i]; d = ei[E + i]; } else { s = d = i - E; }
  const float v0 = lrelu(a_src[s * 2]     + a_dst[d * 2]);
  const float v1 = lrelu(a_src[s * 2 + 1] + a_dst[d * 2 + 1]);
  atomicAdd(&denom[d * 2],     __expf(v0 - e_max[d * 2]));
  atomicAdd(&denom[d * 2 + 1], __expf(v1 - e_max[d * 2 + 1]));
}

// ---------------------------------------------------------------------------
// K5: alpha = exp(e-max)/denom; out[d] += 0.5*(a0*h[s,head0] + a1*h[s,head1]).
// One wave per (edge or self-loop); lane c covers channels c and c+32.
// ---------------------------------------------------------------------------
__global__ __launch_bounds__(256) void gat_scatter(const int* __restrict__ ei, int E, int N,
                                                   const float* __restrict__ a_src,
                                                   const float* __restrict__ a_dst,
                                                   const float* __restrict__ e_max,
                                                   const float* __restrict__ denom,
                                                   const float* __restrict__ h,
                                                   float* __restrict__ out) {
  const int lane = threadIdx.x & 31;
  const long long i = (long long)blockIdx.x * 8 + (threadIdx.x >> 5);
  if (i >= (long long)E + N) return;
  int s, d;
  if (i < E) { s = ei[i]; d = ei[E + i]; } else { s = d = (int)(i - E); }

  const float v0  = lrelu(a_src[s * 2]     + a_dst[d * 2]);
  const float v1  = lrelu(a_src[s * 2 + 1] + a_dst[d * 2 + 1]);
  const float al0 = __expf(v0 - e_max[d * 2])     / denom[d * 2];
  const float al1 = __expf(v1 - e_max[d * 2 + 1]) / denom[d * 2 + 1];

  const float* hs = h + (size_t)s * 128;
  const float m0 = 0.5f * (al0 * hs[lane]      + al1 * hs[64 + lane]);
  const float m1 = 0.5f * (al0 * hs[32 + lane] + al1 * hs[96 + lane]);
  atomicAdd(&out[(size_t)d * 64 + lane],      m0);
  atomicAdd(&out[(size_t)d * 64 + 32 + lane], m1);
}

// ---------------------------------------------------------------------------
extern "C" void kernel_launch(void* const* d_in, const int* in_sizes, int n_in,
                              void* d_out, int out_size, void* d_ws, size_t ws_size,
                              hipStream_t stream) {
  const float* x       = (const float*)d_in[0];
  const float* W       = (const float*)d_in[1];
  const float* att_src = (const float*)d_in[2];
  const float* att_dst = (const float*)d_in[3];
  const float* bias    = (const float*)d_in[4];
  const int*   ei      = (const int*)d_in[5];  // int32 (JAX x64 disabled)

  const int N = in_sizes[0] / 128;   // 50000
  const int E = in_sizes[5] / 2;     // 800000

  // workspace: h[N*128] | a_src[N*2] | a_dst[N*2] | e_max[N*2] | denom[N*2]
  float* h     = (float*)d_ws;
  float* a_src = h + (size_t)N * 128;
  float* a_dst = a_src + (size_t)N * 2;
  float* e_max = a_dst + (size_t)N * 2;
  float* denom = e_max + (size_t)N * 2;
  float* out   = (float*)d_out;

  gat_gemm<<<N / 16, 256, 0, stream>>>(x, W, h);               // N % 16 == 0
  gat_node<<<(N + 7) / 8, 256, 0, stream>>>(h, att_src, att_dst, bias,
                                            a_src, a_dst, e_max, denom, out, N);
  gat_edge_max<<<(E + 255) / 256, 256, 0, stream>>>(ei, E, a_src, a_dst, e_max);
  gat_edge_sum<<<(E + N + 255) / 256, 256, 0, stream>>>(ei, E, N, a_src, a_dst,
                                                        e_max, denom);
  gat_scatter<<<(E + N + 7) / 8, 256, 0, stream>>>(ei, E, N, a_src, a_dst,
                                                   e_max, denom, h, out);
}